// TensorCPField_70884140253839
// MI455X (gfx1250) — compile-verified
//
#include <hip/hip_runtime.h>
#include <hip/hip_bf16.h>

#define RANK   32
#define XS     512
#define YS     512
#define TS     64
#define FEAT   256
#define NTOT   131072
#define NT     256      // samples per block (halves table-staging traffic vs 128)
#define THREADS 512     // 16 waves -> 4 waves/SIMD32 for gather latency hiding
#define FP     (RANK + 1)   // padded feat row stride (odd stride -> conflict-free)

typedef __attribute__((ext_vector_type(2))) float v2f;
typedef __attribute__((ext_vector_type(4))) float v4f;
typedef __attribute__((ext_vector_type(8))) float v8f;

__device__ __forceinline__ int clampi(int v, int lo, int hi) {
    return v < lo ? lo : (v > hi ? hi : v);
}

__launch_bounds__(THREADS, 1)
__global__ void cp_field_fused_kernel(const float* __restrict__ x_idx,
                                      const float* __restrict__ y_idx,
                                      const float* __restrict__ t_idx,
                                      const float* __restrict__ A,
                                      const float* __restrict__ B,
                                      const float* __restrict__ C,
                                      const float* __restrict__ W,
                                      const float* __restrict__ bias,
                                      float* __restrict__ out)
{
    // CDNA5: 320 KB LDS per WGP -> hold every table resident (~203 KB total).
    __shared__ float sA[RANK * XS];        // 64 KB
    __shared__ float sB[RANK * YS];        // 64 KB
    __shared__ float sC[RANK * TS];        //  8 KB
    __shared__ float sW[FEAT * RANK];      // 32 KB  (W[f][r])
    __shared__ float sBias[FEAT];          //  1 KB
    __shared__ float sFeat[NT * FP];       // ~34 KB

    const int tid = threadIdx.x;
    const int n0  = blockIdx.x * NT;

    // ---- Stage 0: cooperative staging of tables into LDS (float4 wide) ----
    {
        const v4f* gA = (const v4f*)A;  v4f* lA = (v4f*)sA;
        for (int i = tid; i < RANK * XS / 4; i += THREADS) lA[i] = gA[i];
        const v4f* gB = (const v4f*)B;  v4f* lB = (v4f*)sB;
        for (int i = tid; i < RANK * YS / 4; i += THREADS) lB[i] = gB[i];
        const v4f* gC = (const v4f*)C;  v4f* lC = (v4f*)sC;
        for (int i = tid; i < RANK * TS / 4; i += THREADS) lC[i] = gC[i];
        const v4f* gW = (const v4f*)W;  v4f* lW = (v4f*)sW;
        for (int i = tid; i < FEAT * RANK / 4; i += THREADS) lW[i] = gW[i];
        if (tid < FEAT) sBias[tid] = bias[tid];
    }
    __syncthreads();

    // ---- Stage 1: gather + 3-way Hadamard contraction over rank i ----
    // thread -> (n_local = tid & 255, r in [16*(tid>>8), +16))
    {
        const int nl = tid & (NT - 1);
        const int rh = tid >> 8;           // 0 or 1
        const int n  = n0 + nl;            // lanes walk consecutive n -> coalesced idx loads
        #pragma unroll 4
        for (int rr = 0; rr < 16; ++rr) {
            const int r = rh * 16 + rr;
            const float xf = x_idx[r * NTOT + n];
            const float yf = y_idx[r * NTOT + n];
            const float tf = t_idx[r * NTOT + n];
            const int xi = clampi((int)(xf * 511.0f),         0, XS - 1);
            const int yi = clampi((int)(yf * 512.0f - 1.0f),  0, YS - 1);
            const int ti = clampi((int)(tf * 64.0f  - 1.0f),  0, TS - 1);
            float acc = 0.0f;
            #pragma unroll
            for (int i = 0; i < RANK; ++i) {
                // row-major [i*stride + idx]: bank = idx & 63, randomized per lane
                acc += sA[i * XS + xi] * sB[i * YS + yi] * sC[i * TS + ti];
            }
            sFeat[nl * FP + r] = acc;
        }
    }
    __syncthreads();

    // ---- Stage 2: out[n0:n0+256, :] = feat(256x32) @ W^T(32x256) + bias ----
    // Full-precision f32 WMMA: D(16x16) = A(16x4) x B(4x16) + C, 8 chained K-steps.
    {
        const int lane = tid & 31;
        const int wave = tid >> 5;         // 16 waves -> 16 M-tiles of 16 rows
        const int lh   = lane >> 4;        // lane half selects K pair
        const int ll   = lane & 15;
        const int m0   = wave * 16;

        for (int ft = 0; ft < 16; ++ft) {
            const int f0 = ft * 16;
            const int f  = f0 + ll;        // N dim striped across lanes

            v8f acc;
            const float bv = sBias[f];
            #pragma unroll
            for (int v = 0; v < 8; ++v) acc[v] = bv;

            #pragma unroll
            for (int k = 0; k < 8; ++k) {
                const int k0 = k * 4;
                // A 16x4 f32 layout: lanes 0-15 -> K={k0,k0+1}, lanes 16-31 -> K={k0+2,k0+3}
                v2f afrag, bfrag;
                afrag.x = sFeat[(m0 + ll) * FP + k0 + 2 * lh + 0];
                afrag.y = sFeat[(m0 + ll) * FP + k0 + 2 * lh + 1];
                // B 4x16 f32: element (K=r, N=f) = W[f][r]
                bfrag.x = sW[f * RANK + k0 + 2 * lh + 0];
                bfrag.y = sW[f * RANK + k0 + 2 * lh + 1];
                acc = __builtin_amdgcn_wmma_f32_16x16x4_f32(
                    /*neg_a=*/false, afrag, /*neg_b=*/false, bfrag,
                    /*c_mod=*/(short)0, acc, /*reuse_a=*/false, /*reuse_b=*/false);
            }

            // C/D layout: VGPR v, lane half h -> M = v + 8h; N = ll
            float* op = out + (size_t)(n0 + m0) * FEAT + f0 + ll;
            #pragma unroll
            for (int v = 0; v < 8; ++v) {
                const int m = v + 8 * lh;
                op[(size_t)m * FEAT] = acc[v];
            }
        }
    }
}

extern "C" void kernel_launch(void* const* d_in, const int* in_sizes, int n_in,
                              void* d_out, int out_size, void* d_ws, size_t ws_size,
                              hipStream_t stream) {
    const float* x_idx = (const float*)d_in[0];
    const float* y_idx = (const float*)d_in[1];
    const float* t_idx = (const float*)d_in[2];
    const float* A     = (const float*)d_in[3];
    const float* B     = (const float*)d_in[4];
    const float* C     = (const float*)d_in[5];
    const float* W     = (const float*)d_in[6];
    const float* bias  = (const float*)d_in[7];
    float* out = (float*)d_out;

    dim3 grid(NTOT / NT);
    dim3 block(THREADS);
    cp_field_fused_kernel<<<grid, block, 0, stream>>>(
        x_idx, y_idx, t_idx, A, B, C, W, bias, out);
}